// DIAF_78761110274129
// MI455X (gfx1250) — compile-verified
//
#include <hip/hip_runtime.h>

typedef float v2f __attribute__((ext_vector_type(2)));
typedef float v4f __attribute__((ext_vector_type(4)));
typedef float v8f __attribute__((ext_vector_type(8)));

#define NUM_T 16

// One thread = one (batch, track). One wave = 32 tracks. 256 threads/block = 8 waves.
// pP = F P F^T computed with V_WMMA_F32_16X16X4_F32 (2 WMMAs per 4 tracks),
// everything else scalar f32 VALU (kernel is HBM-bound: ~189MB @ 23.3TB/s ~ 8us).
//
// All LDS traffic is wave-private and CDNA5 DS ops are in-order per wave, so no
// hardware barriers are needed; __builtin_amdgcn_wave_barrier() is a zero-cost
// compiler scheduling fence keeping the staging / WMMA / readback phases ordered.
__global__ __launch_bounds__(256) void kf_wmma_kernel(
    const float* __restrict__ value,
    const float* __restrict__ vel,
    const float* __restrict__ Pmat,
    const float* __restrict__ obs,
    const int*   __restrict__ lm,
    const int*   __restrict__ mask,
    float* __restrict__ out,
    int Btot, int total_tracks)
{
    // stride-17 padding -> conflict-free for both row-major per-lane access
    // and the WMMA B-operand gather pattern.
    __shared__ float P_lds [256][17];
    __shared__ float pP_lds[256][17];
    __shared__ float G_lds [8][4][16];   // per-wave scratch: (F*P) for current 4-track group

    const int tid   = threadIdx.x;
    const int wave  = tid >> 5;
    const int lane  = tid & 31;
    const int wbase = wave << 5;
    const int T     = blockIdx.x * 256 + tid;

    // wave-uniform tail guard: keeps EXEC all-ones inside live waves (WMMA requirement)
    if ((blockIdx.x * 256 + wbase) >= total_tracks) return;

    const int b = T >> 4;
    const int n = T & 15;

    // ---- global loads (coalesced, single-use -> non-temporal) ----
    v2f val = __builtin_nontemporal_load((const v2f*)(value + (size_t)b*2*NUM_T + 2*n));
    v2f ve  = __builtin_nontemporal_load((const v2f*)(vel   + (size_t)b*2*NUM_T + 2*n));
    v2f ob  = __builtin_nontemporal_load((const v2f*)(obs   + (size_t)b*2*NUM_T + 2*n));
    int lmv = lm  [(size_t)b*NUM_T + n];
    int mkv = mask[(size_t)b*NUM_T + n];

    const v4f* Pp = (const v4f*)(Pmat + (size_t)b*NUM_T*16 + (size_t)n*16);
    v4f p0 = __builtin_nontemporal_load(Pp + 0);
    v4f p1 = __builtin_nontemporal_load(Pp + 1);
    v4f p2 = __builtin_nontemporal_load(Pp + 2);
    v4f p3 = __builtin_nontemporal_load(Pp + 3);

    // stage P row-major into LDS (wave-private region)
    #pragma unroll
    for (int k = 0; k < 4; ++k) {
        P_lds[tid][ 0 + k] = p0[k];
        P_lds[tid][ 4 + k] = p1[k];
        P_lds[tid][ 8 + k] = p2[k];
        P_lds[tid][12 + k] = p3[k];
    }
    __builtin_amdgcn_wave_barrier();

    // ---- constant A operand: A[16x4], row r = F[r&3]; F[r][k] = (k==r)||(k==r+2)
    // A layout: lanes 0-15 hold K=0,1; lanes 16-31 hold K=2,3 (VGPR0=K even-half, VGPR1=K odd)
    const int r4 = lane & 3;
    const int kh = (lane < 16) ? 0 : 2;
    v2f aF;
    aF.x = ((kh     == r4) || (kh     == r4 + 2)) ? 1.0f : 0.0f;
    aF.y = ((kh + 1 == r4) || (kh + 1 == r4 + 2)) ? 1.0f : 0.0f;

    const int nn = lane & 15;
    const int mp = nn >> 2;   // which of the 4 tracks in the group
    const int jj = nn & 3;    // column within the 4x4

    #pragma unroll
    for (int g = 0; g < 8; ++g) {
        const int t = wbase + (g << 2) + mp;

        // B1[k][4*mp+jj] = P_t[k][jj]  (column-blocks of the 4 tracks' P)
        v2f b1;
        b1.x = P_lds[t][(kh    ) * 4 + jj];
        b1.y = P_lds[t][(kh + 1) * 4 + jj];

        v8f zero = {};
        // C1 rows 0-3 (VGPRs 0-3; rows repeat with period 4) = F * P_t
        v8f c1 = __builtin_amdgcn_wmma_f32_16x16x4_f32(
            false, aF, false, b1, (short)0, zero, false, false);

        // lanes 16+n hold exactly the same c1[0..3] as lanes n (C rows 8-11 == rows 0-3),
        // so all 32 lanes store identical values to identical addresses: no EXEC juggling.
        G_lds[wave][mp][ 0 + jj] = c1[0];
        G_lds[wave][mp][ 4 + jj] = c1[1];
        G_lds[wave][mp][ 8 + jj] = c1[2];
        G_lds[wave][mp][12 + jj] = c1[3];
        __builtin_amdgcn_wave_barrier();

        // B2[k][4*mp+jj] = G[jj][k]  => C2 = F * G^T = F P F^T  (P symmetric)
        v2f b2;
        b2.x = G_lds[wave][mp][jj * 4 + kh    ];
        b2.y = G_lds[wave][mp][jj * 4 + kh + 1];

        v8f c2 = __builtin_amdgcn_wmma_f32_16x16x4_f32(
            false, aF, false, b2, (short)0, zero, false, false);

        pP_lds[t][ 0 + jj] = c2[0];
        pP_lds[t][ 4 + jj] = c2[1];
        pP_lds[t][ 8 + jj] = c2[2];
        pP_lds[t][12 + jj] = c2[3];
        __builtin_amdgcn_wave_barrier();
    }

    // ---- per-lane scalar Kalman update ----
    float pP[4][4];
    #pragma unroll
    for (int i = 0; i < 4; ++i)
        #pragma unroll
        for (int j = 0; j < 4; ++j)
            pP[i][j] = pP_lds[tid][i * 4 + j];

    const float predv[4] = { val.x + ve.x, val.y + ve.y, ve.x, ve.y };

    const float R = 1e-5f;
    float S00 = pP[0][0] + R, S01 = pP[0][1];
    float S10 = pP[1][0],     S11 = pP[1][1] + R;
    float det = S00 * S11 - S01 * S10;
    float Si00 =  S11 / det, Si01 = -S01 / det;
    float Si10 = -S10 / det, Si11 =  S00 / det;

    float K0[4], K1[4];
    #pragma unroll
    for (int i = 0; i < 4; ++i) {
        K0[i] = pP[i][0] * Si00 + pP[i][1] * Si10;
        K1[i] = pP[i][0] * Si01 + pP[i][1] * Si11;
    }
    float y0 = ob.x - predv[0], y1 = ob.y - predv[1];

    float upd[4];
    #pragma unroll
    for (int i = 0; i < 4; ++i)
        upd[i] = predv[i] + K0[i] * y0 + K1[i] * y1;

    float Pn[4][4];
    #pragma unroll
    for (int i = 0; i < 4; ++i)
        #pragma unroll
        for (int j = 0; j < 4; ++j)
            Pn[i][j] = pP[i][j] - (K0[i] * pP[0][j] + K1[i] * pP[1][j]);

    const bool zl  = (lmv == 0), zm = (mkv == 0);
    const bool C1c = zl && zm;
    const bool C2c = zl && !zm;
    const bool C3c = !zl && zm;
    const bool C4c = !zl && !zm;
    const bool C3a = C3c && (lmv <= 5);

    float rv0  = C4c ? upd[0] : (C3c ? predv[0] : ob.x);
    float rv1  = C4c ? upd[1] : (C3c ? predv[1] : ob.y);
    float rve0 = C4c ? upd[2] : (C3a ? predv[2] : 0.0f);
    float rve1 = C4c ? upd[3] : (C3a ? predv[3] : 0.0f);
    float rmv  = (C2c || C4c) ? 1.0f : (C3a ? (float)lmv + 1.0f : 0.0f);

    // output tuple flat: rv[B*32] | rm[B*16] | rvel[B*32] | rP[B*256]
    const size_t OFF_RM  = (size_t)Btot * 32;
    const size_t OFF_RVE = (size_t)Btot * 48;
    const size_t OFF_RP  = (size_t)Btot * 80;

    v2f rv = { rv0, rv1 };
    __builtin_nontemporal_store(rv,  (v2f*)(out + (size_t)b*32 + 2*n));
    __builtin_nontemporal_store(rmv, out + OFF_RM + (size_t)b*16 + n);
    v2f rve = { rve0, rve1 };
    __builtin_nontemporal_store(rve, (v2f*)(out + OFF_RVE + (size_t)b*32 + 2*n));

    float* rp = out + OFF_RP + (size_t)b*256 + (size_t)n*16;
    #pragma unroll
    for (int i = 0; i < 4; ++i) {
        v4f row;
        #pragma unroll
        for (int j = 0; j < 4; ++j) {
            float d = (i == j) ? ((i < 2) ? 1.0f : 10.0f) : 0.0f;
            row[j] = C1c ? d : (C3a ? pP[i][j] : (C4c ? Pn[i][j] : 0.0f));
        }
        __builtin_nontemporal_store(row, (v4f*)(rp + i * 4));
    }
}

extern "C" void kernel_launch(void* const* d_in, const int* in_sizes, int n_in,
                              void* d_out, int out_size, void* d_ws, size_t ws_size,
                              hipStream_t stream) {
    const float* value = (const float*)d_in[0];
    const float* vel   = (const float*)d_in[1];
    const float* P     = (const float*)d_in[2];
    const float* obs   = (const float*)d_in[3];
    const int*   lm    = (const int*)d_in[4];
    const int*   mask  = (const int*)d_in[5];

    const int Btot  = in_sizes[0] / 32;       // value is (B, 32)
    const int total = Btot * NUM_T;           // 1,048,576 tracks for B=65536
    const int blocks = (total + 255) / 256;   // 4096 full blocks (exact)

    kf_wmma_kernel<<<blocks, 256, 0, stream>>>(
        value, vel, P, obs, lm, mask, (float*)d_out, Btot, total);
}